// WIN3d_3100966388227
// MI455X (gfx1250) — compile-verified
//
#include <hip/hip_runtime.h>
#include <hip/hip_bf16.h>
#include <stdint.h>

// WIN3d: 3x3x3 sliding-window normalization, [N,C,T,H,W] = [2,32,32,128,128] fp32.
// Memory-bound stencil: separable box sums, async global->LDS tile staging (CDNA5
// GLOBAL_LOAD_ASYNC_TO_LDS_B128 + s_wait_asynccnt), rolling register window over T,
// non-temporal b128 output stores (keep L2 for input halo reuse).

#define TH 8    // output H rows per block
#define TT 16   // output T planes per block
#define RS 132  // padded LDS row stride in floats (528 B, 16B aligned)

typedef float v4f __attribute__((ext_vector_type(4)));   // native vector for NT store

__global__ __launch_bounds__(256) void win3d_kernel(
    const float* __restrict__ x, const float* __restrict__ weight,
    const float* __restrict__ bias, float* __restrict__ out)
{
    constexpr int C = 32, T = 32, H = 128, W = 128;
    constexpr float inv27 = 1.0f / 27.0f;
    constexpr float eps = 1e-5f;

    // raw: double-buffered async-load target, 10 rows (8 + 2 H-halo) of one plane
    __shared__ __align__(16) float raw[2][10][RS];
    __shared__ __align__(16) float wS[10][RS];   // 3-tap W sums
    __shared__ __align__(16) float wQ[10][RS];   // 3-tap W sums of squares

    const int lane = threadIdx.x;      // 0..31
    const int wave = threadIdx.y;      // 0..7  -> output row h0+wave
    const int w0   = lane * 4;         // 4 consecutive w per lane
    const int h0   = blockIdx.x * TH;
    const int t0   = blockIdx.y * TT;
    const int nc   = blockIdx.z;       // n*C + c
    const int c    = nc % C;

    const float wgt = weight[c];
    const float bia = bias[c];

    const size_t base = (size_t)nc * T * H * W;
    const float* xin = x + base;
    float* dst = out + base;

    // Row indices i in [0,9] map to h = h0 + i - 1. Wave v owns row v+1;
    // waves 0 and 7 additionally own the H-halo rows 0 and 9.
    int rows[2]; int nrows = 1;
    rows[0] = wave + 1;
    if (wave == 0) { rows[1] = 0; nrows = 2; }
    if (wave == 7) { rows[1] = 9; nrows = 2; }
    int hg[2];
    for (int r = 0; r < nrows; ++r) {
        int h = h0 + rows[r] - 1;
        hg[r] = (h < 0) ? -h : (h >= H ? 2 * H - 2 - h : h);   // reflect
    }

    // W-halo LDS columns with reflect at tensor edges
    const int wl = (w0 == 0) ? 1 : (w0 - 1);
    const int wr = (w0 + 4 >= W) ? (W - 2) : (w0 + 4);

    float4 Ps2 = {0,0,0,0}, Pq2 = {0,0,0,0};   // 2D box sums, plane t-1
    float4 Ps1 = {0,0,0,0}, Pq1 = {0,0,0,0};   // 2D box sums, plane t
    float4 xc1 = {0,0,0,0};                    // center x, plane t

    for (int k = 0; k < TT + 2; ++k) {
        const int tp  = t0 - 1 + k;
        const int tpR = (tp < 0) ? -tp : (tp >= T ? 2 * T - 2 - tp : tp); // reflect
        const int buf = k & 1;

        // ---- Phase 1: async DMA rows of plane tpR into LDS (16B per lane) ----
        for (int r = 0; r < nrows; ++r) {
            const float* gp = xin + ((size_t)tpR * H + hg[r]) * W + w0;
            unsigned ldsoff = (unsigned)(size_t)&raw[buf][rows[r]][w0];
            asm volatile("global_load_async_to_lds_b128 %0, %1, off"
                         :: "v"(ldsoff), "v"(gp) : "memory");
        }
        // Only this wave reads its own rows below -> asynccnt wait suffices.
        // Inline asm with memory clobber: also a compiler barrier so the LDS
        // reads below cannot be scheduled before the wait.
        asm volatile("s_wait_asynccnt 0" ::: "memory");

        // ---- Phase 2: 3-tap sums along W (in registers) ----
        float4 xcC = {0,0,0,0};
        for (int r = 0; r < nrows; ++r) {
            const int i = rows[r];
            float4 v = *(const float4*)&raw[buf][i][w0];
            float a0 = raw[buf][i][wl];
            float a5 = raw[buf][i][wr];
            float a1 = v.x, a2 = v.y, a3 = v.z, a4 = v.w;
            float4 s, q;
            s.x = a0 + a1 + a2;  s.y = a1 + a2 + a3;
            s.z = a2 + a3 + a4;  s.w = a3 + a4 + a5;
            q.x = a0*a0 + a1*a1 + a2*a2;  q.y = a1*a1 + a2*a2 + a3*a3;
            q.z = a2*a2 + a3*a3 + a4*a4;  q.w = a3*a3 + a4*a4 + a5*a5;
            *(float4*)&wS[i][w0] = s;
            *(float4*)&wQ[i][w0] = q;
            if (i == wave + 1) xcC = v;            // center x of my output row
        }
        __syncthreads();

        // ---- Phase 3: 3-tap sums along H -> 2D box sums of plane tpR ----
        float4 s0 = *(const float4*)&wS[wave    ][w0];
        float4 s1 = *(const float4*)&wS[wave + 1][w0];
        float4 s2 = *(const float4*)&wS[wave + 2][w0];
        float4 q0 = *(const float4*)&wQ[wave    ][w0];
        float4 q1 = *(const float4*)&wQ[wave + 1][w0];
        float4 q2 = *(const float4*)&wQ[wave + 2][w0];
        float4 PsC, PqC;
        PsC.x = s0.x + s1.x + s2.x;  PsC.y = s0.y + s1.y + s2.y;
        PsC.z = s0.z + s1.z + s2.z;  PsC.w = s0.w + s1.w + s2.w;
        PqC.x = q0.x + q1.x + q2.x;  PqC.y = q0.y + q1.y + q2.y;
        PqC.z = q0.z + q1.z + q2.z;  PqC.w = q0.w + q1.w + q2.w;
        __syncthreads();   // protect wS/wQ before next iteration overwrites

        // ---- Phase 4: emit output plane t = tp-1 (needs P(t-1),P(t),P(t+1)) ----
        if (k >= 2) {
            const int t = tp - 1;
            float S, Q, mean, var;
            v4f* op = (v4f*)&dst[((size_t)t * H + (h0 + wave)) * W + w0];
            v4f ov;
            S = Ps2.x + Ps1.x + PsC.x;  Q = Pq2.x + Pq1.x + PqC.x;
            mean = S * inv27;  var = Q * inv27 - mean * mean;
            ov.x = (xc1.x - mean) * rsqrtf(var + eps) * wgt + bia;
            S = Ps2.y + Ps1.y + PsC.y;  Q = Pq2.y + Pq1.y + PqC.y;
            mean = S * inv27;  var = Q * inv27 - mean * mean;
            ov.y = (xc1.y - mean) * rsqrtf(var + eps) * wgt + bia;
            S = Ps2.z + Ps1.z + PsC.z;  Q = Pq2.z + Pq1.z + PqC.z;
            mean = S * inv27;  var = Q * inv27 - mean * mean;
            ov.z = (xc1.z - mean) * rsqrtf(var + eps) * wgt + bia;
            S = Ps2.w + Ps1.w + PsC.w;  Q = Pq2.w + Pq1.w + PqC.w;
            mean = S * inv27;  var = Q * inv27 - mean * mean;
            ov.w = (xc1.w - mean) * rsqrtf(var + eps) * wgt + bia;
            // Output is write-once: non-temporal store, keep L2 for input halos.
            __builtin_nontemporal_store(ov, op);
        }

        // roll the T window
        Ps2 = Ps1; Pq2 = Pq1;
        Ps1 = PsC; Pq1 = PqC;
        xc1 = xcC;
    }
}

extern "C" void kernel_launch(void* const* d_in, const int* in_sizes, int n_in,
                              void* d_out, int out_size, void* d_ws, size_t ws_size,
                              hipStream_t stream) {
    (void)in_sizes; (void)n_in; (void)d_ws; (void)ws_size; (void)out_size;
    const float* x      = (const float*)d_in[0];
    const float* weight = (const float*)d_in[1];
    const float* bias   = (const float*)d_in[2];
    float* out = (float*)d_out;

    // N=2, C=32, T=32, H=128, W=128
    dim3 grid(128 / TH, 32 / TT, 2 * 32);   // (H tiles, T tiles, N*C)
    dim3 block(32, 8, 1);                   // 8 wave32 waves
    hipLaunchKernelGGL(win3d_kernel, grid, block, 0, stream, x, weight, bias, out);
}